// Rnn_Model1_77326591197593
// MI455X (gfx1250) — compile-verified
//
#include <hip/hip_runtime.h>
#include <math.h>

#define T_STEPS 32768
#define D_IN    64
#define H_DIM   128
#define G_DIM   512   // 4*H
#define NOUT    15    // 11 + 2 + 1 + 1 head outputs

typedef float v2f __attribute__((ext_vector_type(2)));
typedef float v8f __attribute__((ext_vector_type(8)));

// ---------------------------------------------------------------------------
// Kernel 1: Xg[t][g] = X[t,:] . W_ih[g,:] + b_ih[g] + b_hh[g]
// (32768 x 64) @ (64 x 512) via V_WMMA_F32_16X16X4_F32, one 16x16 tile/wave,
// K=64 chained as 16 WMMA ops accumulating in the 8-VGPR f32 accumulator.
// ---------------------------------------------------------------------------
__global__ __launch_bounds__(128) void lstm_input_proj(
    const float* __restrict__ X, const float* __restrict__ W_ih,
    const float* __restrict__ b_ih, const float* __restrict__ b_hh,
    float* __restrict__ Xg)
{
  const int lane  = threadIdx.x & 31;
  const int wave  = threadIdx.x >> 5;
  const int tile  = blockIdx.x * 4 + wave;
  const int tileT = tile >> 5;       // 2048 tiles of 16 timesteps
  const int tileG = tile & 31;       // 32 tiles of 16 gates
  const int half  = lane >> 4;       // 0 = lanes 0-15, 1 = lanes 16-31
  const int l15   = lane & 15;

  const int row = tileT * 16 + l15;  // A-matrix M index (timestep)
  const int col = tileG * 16 + l15;  // B-matrix N index (gate)

  v8f c = {};
  #pragma unroll
  for (int k0 = 0; k0 < D_IN; k0 += 4) {
    // ISA 7.12.2: 32-bit A 16x4 -> v0 holds K={k0, k0+2}, v1 holds K={k0+1, k0+3}
    const int ka = k0 + half * 2;
    v2f a, b;
    a.x = X[row * D_IN + ka];
    a.y = X[row * D_IN + ka + 1];
    // B (4x16) mirrors A's K striping; B row = K, striped over lanes by N
    b.x = W_ih[col * D_IN + ka];
    b.y = W_ih[col * D_IN + ka + 1];
    c = __builtin_amdgcn_wmma_f32_16x16x4_f32(false, a, false, b,
                                              (short)0, c, false, false);
  }

  const float bias = b_ih[col] + b_hh[col];
  // C/D layout: VGPR j holds M = j (lanes 0-15) / j+8 (lanes 16-31), N = l15
  #pragma unroll
  for (int j = 0; j < 8; ++j) {
    const int t = tileT * 16 + j + half * 8;
    Xg[(size_t)t * G_DIM + col] = c[j] + bias;
  }
}

// ---------------------------------------------------------------------------
// Kernel 2: sequential LSTM recurrence. One persistent 512-thread workgroup
// (16 waves = full WGP). Thread g keeps W_hh row g in 128 VGPRs; h/c/gate
// exchange via LDS; Xg prefetched one step ahead (lives in 192MB L2).
// ---------------------------------------------------------------------------
__device__ __forceinline__ float fsigmoid(float x) {
  return 1.0f / (1.0f + __expf(-x));
}
__device__ __forceinline__ float ftanh(float x) {
  float e = __expf(2.0f * x);
  return (e - 1.0f) / (e + 1.0f);
}

__global__ __launch_bounds__(512) void lstm_recurrence(
    const float* __restrict__ W_hh, const float* __restrict__ Xg,
    float* __restrict__ hs)
{
  __shared__ float h_lds[H_DIM];
  __shared__ float gates[G_DIM];
  const int g = threadIdx.x;   // gate row: i[0..127] f[128..255] g[256..383] o[384..511]

  // Register-resident weight row (64 float2 = 128 VGPRs, fully unrolled).
  float2 w[H_DIM / 2];
  #pragma unroll
  for (int j = 0; j < H_DIM / 2; ++j)
    w[j] = *(const float2*)(W_hh + (size_t)g * H_DIM + 2 * j);

  if (g < H_DIM) h_lds[g] = 0.0f;
  float c = 0.0f;
  float xg_cur = Xg[g];
  __syncthreads();

  for (int t = 0; t < T_STEPS; ++t) {
    // Prefetch next step's input-projection value (hides L2 latency).
    float xg_nxt = (t + 1 < T_STEPS) ? Xg[(size_t)(t + 1) * G_DIM + g] : 0.0f;

    // 128-MAC dot: 4 independent accumulators over float2 pairs (ILP / pk-fma).
    const float2* h2 = (const float2*)h_lds;
    float a0 = 0.0f, a1 = 0.0f, a2 = 0.0f, a3 = 0.0f;
    #pragma unroll
    for (int j = 0; j < H_DIM / 2; j += 2) {
      float2 wa = w[j],     ha = h2[j];
      float2 wb = w[j + 1], hb = h2[j + 1];
      a0 = fmaf(wa.x, ha.x, a0);
      a1 = fmaf(wa.y, ha.y, a1);
      a2 = fmaf(wb.x, hb.x, a2);
      a3 = fmaf(wb.y, hb.y, a3);
    }
    gates[g] = xg_cur + ((a0 + a1) + (a2 + a3));
    __syncthreads();                 // all dots done; h_lds free to overwrite

    if (g < H_DIM) {
      float gi = gates[g];
      float gf = gates[H_DIM + g];
      float gg = gates[2 * H_DIM + g];
      float go = gates[3 * H_DIM + g];
      c = fsigmoid(gf) * c + fsigmoid(gi) * ftanh(gg);
      float h = fsigmoid(go) * ftanh(c);
      h_lds[g] = h;
      hs[(size_t)t * H_DIM + g] = h;  // fire-and-forget global store
    }
    __syncthreads();                 // h ready for next step
    xg_cur = xg_nxt;
  }
}

// ---------------------------------------------------------------------------
// Kernel 3: fused output heads. 15 total output dims; weights+biases staged
// in LDS (broadcast reads), one timestep per thread.
// d_out layout: step (T,11) | experience (T,2) | rsd (T,1) | s (T,1)
// ---------------------------------------------------------------------------
__global__ __launch_bounds__(256) void lstm_heads(
    const float* __restrict__ hs,
    const float* __restrict__ W1, const float* __restrict__ b1,
    const float* __restrict__ W2, const float* __restrict__ b2,
    const float* __restrict__ W3, const float* __restrict__ b3,
    const float* __restrict__ W4, const float* __restrict__ b4,
    float* __restrict__ out)
{
  __shared__ float Wl[NOUT * H_DIM];
  __shared__ float bl[NOUT];

  for (int idx = threadIdx.x; idx < NOUT * H_DIM; idx += 256) {
    int r = idx >> 7, cc = idx & (H_DIM - 1);
    float v;
    if (r < 11)      v = W1[r * H_DIM + cc];
    else if (r < 13) v = W2[(r - 11) * H_DIM + cc];
    else if (r == 13) v = W3[cc];
    else              v = W4[cc];
    Wl[idx] = v;
  }
  if (threadIdx.x < NOUT) {
    int r = threadIdx.x;
    bl[r] = (r < 11) ? b1[r] : (r < 13) ? b2[r - 11] : (r == 13) ? b3[0] : b4[0];
  }
  __syncthreads();

  const int t = blockIdx.x * 256 + threadIdx.x;
  float acc[NOUT];
  #pragma unroll
  for (int o = 0; o < NOUT; ++o) acc[o] = bl[o];

  const float* h = hs + (size_t)t * H_DIM;
  for (int k = 0; k < H_DIM; k += 4) {
    float4 hv = *(const float4*)(h + k);
    #pragma unroll
    for (int o = 0; o < NOUT; ++o) {
      acc[o] = fmaf(hv.x, Wl[o * H_DIM + k],     acc[o]);
      acc[o] = fmaf(hv.y, Wl[o * H_DIM + k + 1], acc[o]);
      acc[o] = fmaf(hv.z, Wl[o * H_DIM + k + 2], acc[o]);
      acc[o] = fmaf(hv.w, Wl[o * H_DIM + k + 3], acc[o]);
    }
  }

  #pragma unroll
  for (int o = 0; o < 11; ++o) out[(size_t)t * 11 + o] = acc[o];
  out[(size_t)T_STEPS * 11 + t * 2 + 0] = acc[11];
  out[(size_t)T_STEPS * 11 + t * 2 + 1] = acc[12];
  out[(size_t)T_STEPS * 13 + t]         = acc[13];
  out[(size_t)T_STEPS * 14 + t]         = acc[14];
}

// ---------------------------------------------------------------------------
extern "C" void kernel_launch(void* const* d_in, const int* in_sizes, int n_in,
                              void* d_out, int out_size, void* d_ws, size_t ws_size,
                              hipStream_t stream) {
  const float* X    = (const float*)d_in[0];
  const float* W_ih = (const float*)d_in[1];
  const float* W_hh = (const float*)d_in[2];
  const float* b_ih = (const float*)d_in[3];
  const float* b_hh = (const float*)d_in[4];
  const float* W1   = (const float*)d_in[5];
  const float* b1   = (const float*)d_in[6];
  const float* W2   = (const float*)d_in[7];
  const float* b2   = (const float*)d_in[8];
  const float* W3   = (const float*)d_in[9];
  const float* b3   = (const float*)d_in[10];
  const float* W4   = (const float*)d_in[11];
  const float* b4   = (const float*)d_in[12];
  float* out = (float*)d_out;

  float* Xg = (float*)d_ws;                         // T*512 f32 = 64 MB
  float* hs = Xg + (size_t)T_STEPS * G_DIM;         // T*128 f32 = 16 MB

  // 1) parallel input projection (fp32 WMMA), biases folded in
  lstm_input_proj<<<(T_STEPS / 16) * (G_DIM / 16) / 4, 128, 0, stream>>>(
      X, W_ih, b_ih, b_hh, Xg);

  // 2) sequential recurrence: one persistent workgroup on one WGP
  lstm_recurrence<<<1, 512, 0, stream>>>(W_hh, Xg, hs);

  // 3) output heads
  lstm_heads<<<T_STEPS / 256, 256, 0, stream>>>(
      hs, W1, b1, W2, b2, W3, b3, W4, b4, out);
}